// SimpleGCN_34900904247992
// MI455X (gfx1250) — compile-verified
//
#include <hip/hip_runtime.h>
#include <hip/hip_bf16.h>

typedef __attribute__((ext_vector_type(2))) float v2f;
typedef __attribute__((ext_vector_type(8))) float v8f;
typedef __attribute__((ext_vector_type(4))) int   v4i;
typedef __attribute__((address_space(3))) v4i     lds_v4i;

#define IN_DIM 128

#if defined(__gfx1250__) && __has_builtin(__builtin_amdgcn_global_load_async_to_lds_b128)
#define HAVE_ASYNC_LDS 1
#else
#define HAVE_ASYNC_LDS 0
#endif

// ---------------- degree / normalization ----------------

__global__ void k_init_deg(float* __restrict__ deg, int n) {
    int i = blockIdx.x * blockDim.x + threadIdx.x;
    if (i < n) deg[i] = 1.0f;                 // self-loop contribution
}

__global__ void k_accum_deg(const long long* __restrict__ dstI,
                            float* __restrict__ deg, int E) {
    int e = blockIdx.x * blockDim.x + threadIdx.x;
    if (e < E) atomicAdd(&deg[(int)dstI[e]], 1.0f);
}

__global__ void k_rsqrt_inplace(float* __restrict__ d, int n) {
    int i = blockIdx.x * blockDim.x + threadIdx.x;
    if (i < n) {
        float v = d[i];
        d[i] = (v > 0.0f) ? rsqrtf(v) : 0.0f;
    }
}

// ---------------- fp32 WMMA GEMM: C[M,NC] = A[M,128] * W[128,NC] ----------------
// W (128 x NC) staged into LDS once per block via async global->LDS copies;
// one wave computes a 16x16 output tile; K=128 via 32 x V_WMMA_F32_16X16X4_F32.
// NC is a compile-time constant so all LDS B-reads use immediate ds offsets.

template <int NC>
__global__ void k_gemm_wmma(const float* __restrict__ A, const float* __restrict__ W,
                            float* __restrict__ C, int totalTiles) {
    extern __shared__ float sW[];             // 128 * NC floats

    // ---- stage W into LDS (all threads participate) ----
    constexpr int nVec = (IN_DIM * NC) >> 2;  // float4 chunks
#pragma unroll
    for (int v = threadIdx.x; v < nVec; v += 256) {
#if HAVE_ASYNC_LDS
        __builtin_amdgcn_global_load_async_to_lds_b128(
            (v4i*)(W + 4 * v), (lds_v4i*)(sW + 4 * v), 0, 0);
#else
        ((float4*)sW)[v] = ((const float4*)W)[v];
#endif
    }
#if HAVE_ASYNC_LDS
#if __has_builtin(__builtin_amdgcn_s_wait_asynccnt)
    __builtin_amdgcn_s_wait_asynccnt(0);
#else
    asm volatile("s_wait_asynccnt 0" ::: "memory");
#endif
#endif
    __syncthreads();

    int wave = blockIdx.x * (blockDim.x >> 5) + (threadIdx.x >> 5);
    if (wave >= totalTiles) return;           // wave-uniform; EXEC stays all-ones
    int lane = threadIdx.x & 31;
    int hi   = lane >> 4;                     // 0: K pair {0,1}, 1: K pair {2,3}
    int l15  = lane & 15;

    constexpr int nTiles = NC / 16;           // power of two -> shifts
    int mT = wave / nTiles;
    int nT = wave - mT * nTiles;
    int mBase = mT << 4;
    int nBase = nT << 4;

    const float* Arow = A + (size_t)(mBase + l15) * IN_DIM;  // A-frag row = lane%16
    int col = nBase + l15;                                   // B/C column = lane%16
    const float* Wcol = sW + col;             // per-lane LDS base; rest is imm offsets

    v8f acc = {0.f, 0.f, 0.f, 0.f, 0.f, 0.f, 0.f, 0.f};
#pragma unroll
    for (int ks = 0; ks < 32; ++ks) {
        int kb = (ks << 2) + (hi << 1);       // K base for this half-wave
        v2f a;
        a.x = Arow[kb];
        a.y = Arow[kb + 1];
        v2f b;
        b.x = Wcol[kb * NC];                  // ds_load_b32 ... offset:kb*NC*4
        b.y = Wcol[(kb + 1) * NC];
        acc = __builtin_amdgcn_wmma_f32_16x16x4_f32(false, a, false, b,
                                                    (short)0, acc, false, false);
    }
#pragma unroll
    for (int r = 0; r < 8; ++r) {             // C layout: VGPR r -> row r (+8 for hi half)
        int row = mBase + r + (hi << 3);
        C[(size_t)row * NC + col] = acc[r];
    }
}

// ---------------- aggregation ----------------

// out[i,f] = dinv[i]^2 * h[i,f] + bias[f]   (self-loop + bias seed)
__global__ void k_seed(const float* __restrict__ h, const float* __restrict__ dinv,
                       const float* __restrict__ bias, float* __restrict__ out,
                       int n, int dimShift, int dimMask) {
    int t = blockIdx.x * blockDim.x + threadIdx.x;
    int i = t >> dimShift;
    if (i >= n) return;
    int f = t & dimMask;
    float di = dinv[i];
    out[t] = di * di * h[t] + bias[f];
}

// out[dst,f..f+3] += dinv[src]*dinv[dst] * h[src,f..f+3]
__global__ void k_scatter(const long long* __restrict__ srcI,
                          const long long* __restrict__ dstI,
                          const float* __restrict__ dinv,
                          const float* __restrict__ h, float* __restrict__ out,
                          int E, int dim, int tpeShift, int tpeMask) {
    int t = blockIdx.x * blockDim.x + threadIdx.x;
    int e = t >> tpeShift;
    if (e >= E) return;
    int f = (t & tpeMask) << 2;
    int s = (int)srcI[e];
    int d = (int)dstI[e];
    float nrm = dinv[s] * dinv[d];
    const float4 hv = *(const float4*)(h + (size_t)s * dim + f);
    float* o = out + (size_t)d * dim + f;
    atomicAdd(o + 0, nrm * hv.x);
    atomicAdd(o + 1, nrm * hv.y);
    atomicAdd(o + 2, nrm * hv.z);
    atomicAdd(o + 3, nrm * hv.w);
}

__global__ void k_relu(float* __restrict__ x, long n) {
    long i = (long)blockIdx.x * blockDim.x + threadIdx.x;
    if (i < n) x[i] = fmaxf(x[i], 0.0f);
}

// ---------------- launch ----------------

extern "C" void kernel_launch(void* const* d_in, const int* in_sizes, int n_in,
                              void* d_out, int out_size, void* d_ws, size_t ws_size,
                              hipStream_t stream) {
    const float*     x   = (const float*)d_in[0];
    const float*     W1  = (const float*)d_in[1];
    const float*     b1  = (const float*)d_in[2];
    const float*     W2  = (const float*)d_in[3];
    const float*     b2  = (const float*)d_in[4];
    const long long* ei  = (const long long*)d_in[5];

    const int HID = in_sizes[2];              // 128
    const int OUT = in_sizes[4];              // 64
    const int N   = in_sizes[0] / IN_DIM;     // 100000
    const int E   = in_sizes[5] / 2;          // 1600000
    const long long* srcI = ei;
    const long long* dstI = ei + E;

    float* out  = (float*)d_out;
    float* dinv = (float*)d_ws;
    float* h1   = dinv + N;                   // N x 128
    float* agg1 = h1 + (size_t)N * HID;       // N x 128
    float* h2   = h1;                         // reuse h1 buffer for N x 64

    const int T = 256;
    // 1. normalization vector
    k_init_deg<<<(N + T - 1) / T, T, 0, stream>>>(dinv, N);
    k_accum_deg<<<(E + T - 1) / T, T, 0, stream>>>(dstI, dinv, E);
    k_rsqrt_inplace<<<(N + T - 1) / T, T, 0, stream>>>(dinv, N);

    // 2. h1 = x @ W1   (tiles: (N/16) x (128/16))
    {
        int totalTiles = (N / 16) * (128 / 16);
        int grid = (totalTiles + (T / 32) - 1) / (T / 32);
        size_t shmem = (size_t)IN_DIM * 128 * sizeof(float);   // 64 KB
        k_gemm_wmma<128><<<grid, T, shmem, stream>>>(x, W1, h1, totalTiles);
    }

    // 3. layer-1 aggregation: seed (self loop + bias), edge scatter, ReLU
    {
        long nt = (long)N * HID;
        k_seed<<<(int)((nt + T - 1) / T), T, 0, stream>>>(h1, dinv, b1, agg1, N, 7, 127);
        long st = (long)E * (HID / 4);
        k_scatter<<<(int)((st + T - 1) / T), T, 0, stream>>>(srcI, dstI, dinv, h1, agg1,
                                                            E, HID, 5, 31);
        k_relu<<<(int)((nt + T - 1) / T), T, 0, stream>>>(agg1, nt);
    }

    // 4. h2 = agg1 @ W2
    {
        int totalTiles = (N / 16) * (64 / 16);
        int grid = (totalTiles + (T / 32) - 1) / (T / 32);
        size_t shmem = (size_t)IN_DIM * 64 * sizeof(float);    // 32 KB
        k_gemm_wmma<64><<<grid, T, shmem, stream>>>(agg1, W2, h2, totalTiles);
    }

    // 5. layer-2 aggregation into d_out
    {
        long nt = (long)N * OUT;
        k_seed<<<(int)((nt + T - 1) / T), T, 0, stream>>>(h2, dinv, b2, out, N, 6, 63);
        long st = (long)E * (OUT / 4);
        k_scatter<<<(int)((st + T - 1) / T), T, 0, stream>>>(srcI, dstI, dinv, h2, out,
                                                            E, OUT, 4, 15);
    }
}